// ResGraphModule_28398323761987
// MI455X (gfx1250) — compile-verified
//
#include <hip/hip_runtime.h>

#define N_NODES 100000
#define N_EDGES 500000
#define D_IN    128
#define D_OUT   128
#define D_EDGE  64
#define EPS_1P  (1.0f + 1e-5f)
#define INV_SQRT2 0.70710678118654752f

typedef __attribute__((ext_vector_type(2))) float v2f;
typedef __attribute__((ext_vector_type(4))) float v4f;
typedef __attribute__((ext_vector_type(8))) float v8f;

__device__ __forceinline__ v8f wmma_f32_k4(v2f a, v2f b, v8f c) {
    // D(16x16 f32) += A(16x4 f32) * B(4x16 f32)
    return __builtin_amdgcn_wmma_f32_16x16x4_f32(
        /*neg_a=*/false, a, /*neg_b=*/false, b,
        /*c_mod=*/(short)0, c, /*reuse_a=*/false, /*reuse_b=*/false);
}

// ---------------------------------------------------------------- zero aggr
__global__ __launch_bounds__(256) void zero_kernel(float* __restrict__ p, int n4) {
    int i = blockIdx.x * blockDim.x + threadIdx.x;
    int stride = gridDim.x * blockDim.x;
    v4f z = {};
    for (; i < n4; i += stride) ((v4f*)p)[i] = z;
}

// ------------------------------------------- edge GEMM + message + scatter
// ea = edge_attr @ W_edge^T ; msg = relu(x[src] + ea) ; aggr[dst] += msg
__global__ __launch_bounds__(256) void edge_kernel(
    const float* __restrict__ x,
    const long long* __restrict__ edge_index,
    const float* __restrict__ edge_attr,
    const float* __restrict__ W_edge,      // [128][64] row-major = [n][k]
    float* __restrict__ aggr)
{
    __shared__ float sW[D_IN * D_EDGE];    // 32 KB weights
    __shared__ float sEA[16 * D_EDGE];     // 4 KB edge_attr tile
    __shared__ int   sSrc[16];
    __shared__ int   sDst[16];

    const int tid  = threadIdx.x;
    const int lane = tid & 31;
    const int wv   = tid >> 5;             // 8 waves -> 8 n-tiles of 16

    // stage W_edge once per block
    for (int i = tid * 4; i < D_IN * D_EDGE; i += 256 * 4)
        *(v4f*)&sW[i] = *(const v4f*)&W_edge[i];

    const int kb     = (lane >> 4) << 1;   // K sub-offset: 0 (lanes 0-15) / 2 (16-31)
    const int nl     = lane & 15;          // A-row M / B-col N within tile
    const int mshift = (lane >> 4) << 3;   // D-row base: 0 / 8
    const int ncol   = wv * 16 + nl;       // output column 0..127

    const int ntiles = N_EDGES / 16;       // 31250, exact
    for (int t = blockIdx.x; t < ntiles; t += gridDim.x) {
        const int e0 = t * 16;
        __syncthreads();                   // previous tile's LDS reads done
        {   // stage 16x64 edge_attr tile: 256 threads x 1 float4
            int m = tid >> 4;
            int k = (tid & 15) * 4;
            // 32-bit offset: max 500000*64 floats = 128 MB < 2^31
            *(v4f*)&sEA[m * D_EDGE + k] =
                *(const v4f*)&edge_attr[(e0 + m) * D_EDGE + k];
        }
        if (tid < 16) {
            sSrc[tid] = (int)edge_index[e0 + tid] * D_IN;          // element offsets
            sDst[tid] = (int)edge_index[(size_t)N_EDGES + e0 + tid] * D_IN;
        }
        __syncthreads();

        // Issue all gather loads up front so they overlap the WMMA chain.
        // 32-bit offsets keep the compiler on SADDR+voffset (GVS) addressing.
        float xg[8];
        int   doff[8];
        #pragma unroll
        for (int r = 0; r < 8; ++r) {
            int m = r + mshift;            // edge row in tile (D-matrix layout)
            xg[r]   = x[sSrc[m] + ncol];
            doff[r] = sDst[m] + ncol;
        }

        // Two independent accumulator chains (even/odd k-steps) for XDL ILP.
        v8f acc0 = {};
        v8f acc1 = {};
        #pragma unroll
        for (int k0 = 0; k0 < D_EDGE; k0 += 8) {
            v2f a0 = *(v2f*)&sEA[nl * D_EDGE + k0 + kb];
            v2f b0 = *(v2f*)&sW[ncol * D_EDGE + k0 + kb];
            v2f a1 = *(v2f*)&sEA[nl * D_EDGE + k0 + 4 + kb];
            v2f b1 = *(v2f*)&sW[ncol * D_EDGE + k0 + 4 + kb];
            acc0 = wmma_f32_k4(a0, b0, acc0);
            acc1 = wmma_f32_k4(a1, b1, acc1);
        }
        v8f acc = acc0 + acc1;

        // msg = relu(x[src] + ea); scatter-add (aggr/x are L2-resident)
        #pragma unroll
        for (int r = 0; r < 8; ++r) {
            float v = acc[r] + xg[r];
            v = v > 0.f ? v : 0.f;
            atomicAdd(aggr + doff[r], v);
        }
    }
}

// --------------------------- node update: two fused WMMA GEMMs + epilogue
// h = relu(((1+eps)x + aggr) @ W_nn^T + b) ; out = (h + x @ W_res^T)/sqrt2
__global__ __launch_bounds__(256) void node_kernel(
    const float* __restrict__ x,
    const float* __restrict__ aggr,
    const float* __restrict__ W_nn,        // [128][128] = [n][k]
    const float* __restrict__ b_nn,
    const float* __restrict__ W_res,       // [128][128] = [n][k]
    float* __restrict__ out)
{
    __shared__ float sU[16 * D_IN];        // (1+eps)*x + aggr tile, 8 KB
    __shared__ float sX[16 * D_IN];        // x tile, 8 KB

    const int tid  = threadIdx.x;
    const int lane = tid & 31;
    const int wv   = tid >> 5;

    const int kb     = (lane >> 4) << 1;
    const int nl     = lane & 15;
    const int mshift = (lane >> 4) << 3;
    const int ncol   = wv * 16 + nl;

    const float bias = b_nn[ncol];

    const int ntiles = N_NODES / 16;       // 6250, exact
    for (int t = blockIdx.x; t < ntiles; t += gridDim.x) {
        const int n0 = t * 16;
        __syncthreads();
        {   // stage x and u tiles: 256 threads x 8 floats each array
            int m = tid >> 4;
            int c = (tid & 15) * 8;
            int g = (n0 + m) * D_IN + c;   // 32-bit: max 12.8M elements
            v4f xa = *(const v4f*)&x[g];
            v4f xb = *(const v4f*)&x[g + 4];
            v4f ga = *(const v4f*)&aggr[g];
            v4f gb = *(const v4f*)&aggr[g + 4];
            *(v4f*)&sX[m * D_IN + c]     = xa;
            *(v4f*)&sX[m * D_IN + c + 4] = xb;
            *(v4f*)&sU[m * D_IN + c]     = xa * EPS_1P + ga;
            *(v4f*)&sU[m * D_IN + c + 4] = xb * EPS_1P + gb;
        }
        __syncthreads();

        // Four independent accumulator chains: {H,R} x {even,odd k-step}.
        v8f accH0 = {}, accH1 = {};
        v8f accR0 = {}, accR1 = {};
        #pragma unroll 4
        for (int k0 = 0; k0 < D_IN; k0 += 8) {
            v2f aH0 = *(v2f*)&sU[nl * D_IN + k0 + kb];
            v2f bH0 = *(const v2f*)&W_nn[ncol * D_IN + k0 + kb];
            v2f aR0 = *(v2f*)&sX[nl * D_IN + k0 + kb];
            v2f bR0 = *(const v2f*)&W_res[ncol * D_IN + k0 + kb];
            v2f aH1 = *(v2f*)&sU[nl * D_IN + k0 + 4 + kb];
            v2f bH1 = *(const v2f*)&W_nn[ncol * D_IN + k0 + 4 + kb];
            v2f aR1 = *(v2f*)&sX[nl * D_IN + k0 + 4 + kb];
            v2f bR1 = *(const v2f*)&W_res[ncol * D_IN + k0 + 4 + kb];
            accH0 = wmma_f32_k4(aH0, bH0, accH0);
            accR0 = wmma_f32_k4(aR0, bR0, accR0);
            accH1 = wmma_f32_k4(aH1, bH1, accH1);
            accR1 = wmma_f32_k4(aR1, bR1, accR1);
        }
        v8f accH = accH0 + accH1;
        v8f accR = accR0 + accR1;

        #pragma unroll
        for (int r = 0; r < 8; ++r) {
            int m = r + mshift;
            float h = accH[r] + bias;
            h = h > 0.f ? h : 0.f;
            out[(n0 + m) * D_OUT + ncol] = (h + accR[r]) * INV_SQRT2;
        }
    }
}

// ---------------------------------------------------------------- launcher
extern "C" void kernel_launch(void* const* d_in, const int* in_sizes, int n_in,
                              void* d_out, int out_size, void* d_ws, size_t ws_size,
                              hipStream_t stream) {
    const float*     x          = (const float*)d_in[0];
    const long long* edge_index = (const long long*)d_in[1];   // int64 per reference
    const float*     edge_attr  = (const float*)d_in[2];
    const float*     W_edge     = (const float*)d_in[3];
    const float*     W_nn       = (const float*)d_in[4];
    const float*     b_nn       = (const float*)d_in[5];
    const float*     W_res      = (const float*)d_in[6];
    float*           out        = (float*)d_out;
    float*           aggr       = (float*)d_ws;                // N*128 f32 = 51.2 MB

    zero_kernel<<<2048, 256, 0, stream>>>(aggr, (N_NODES * D_IN) / 4);
    edge_kernel<<<2048, 256, 0, stream>>>(x, edge_index, edge_attr, W_edge, aggr);
    node_kernel<<<2048, 256, 0, stream>>>(x, aggr, W_nn, b_nn, W_res, out);
}